// GINLayer_47210280517997
// MI455X (gfx1250) — compile-verified
//
#include <hip/hip_runtime.h>

#define NNODES 50000
#define NEDGES 800000
#define DFEAT  128
#define BN_EPS 1e-5f

typedef __attribute__((ext_vector_type(2))) float v2f;
typedef __attribute__((ext_vector_type(8))) float v8f;

// ---------------------------------------------------------------------------
// Kernel 1: aggr = x  (self term of GIN aggregation), zero the stats buffer
// ---------------------------------------------------------------------------
__global__ __launch_bounds__(256)
void gin_init_kernel(const float* __restrict__ x, float* __restrict__ aggr,
                     float* __restrict__ stats, int n4) {
    int i = blockIdx.x * 256 + threadIdx.x;
    if (i < n4) {
        ((float4*)aggr)[i] = ((const float4*)x)[i];
    }
    if (blockIdx.x == 0 && threadIdx.x < 256) {
        stats[threadIdx.x] = 0.0f;
    }
}

// ---------------------------------------------------------------------------
// Kernel 2: edge scatter: aggr[dst] += x[src].  One wave per edge:
// scalar src/dst loads, coalesced 512B float4 gather, 4 fp32 atomics/lane.
// ---------------------------------------------------------------------------
__global__ __launch_bounds__(256)
void gin_scatter_kernel(const float* __restrict__ x,
                        const int* __restrict__ src,
                        const int* __restrict__ dst,
                        float* __restrict__ aggr, int nedges) {
    unsigned tid  = blockIdx.x * 256u + threadIdx.x;
    int      e    = (int)(tid >> 5);
    int      lane = (int)(tid & 31u);
    if (e >= nedges) return;
    int s = src[e];
    int d = dst[e];
    float4 v = ((const float4*)(x + (size_t)s * DFEAT))[lane];
    float* p = aggr + (size_t)d * DFEAT + lane * 4;
    unsafeAtomicAdd(p + 0, v.x);
    unsafeAtomicAdd(p + 1, v.y);
    unsafeAtomicAdd(p + 2, v.z);
    unsafeAtomicAdd(p + 3, v.w);
}

// ---------------------------------------------------------------------------
// Kernel 3: fused MLP: Out = (relu(A @ W1 + b1)) @ W2 + b2, fp32 WMMA.
// One 128-row tile per block (256 threads = 8 waves, 16-row strip per wave).
// Weights are staged TRANSPOSED in LDS so every A/B fragment is a single
// aligned ds_load_b64.  132KB LDS keeps 2 workgroups resident per WGP so the
// inter-GEMM W2 reload of one block overlaps the other block's WMMAs; W2 is
// additionally warmed into cache via global_prefetch (no VGPR cost).  The A
// tile is overwritten in-place by h1 = relu(acc + b1).
// ---------------------------------------------------------------------------
__global__ __launch_bounds__(256)
void gin_mlp_fused_kernel(const float* __restrict__ A,
                          const float* __restrict__ W1,
                          const float* __restrict__ b1,
                          const float* __restrict__ W2,
                          const float* __restrict__ b2,
                          float* __restrict__ Out, int nrows) {
    __shared__ float As[128][132];   // A tile, then h1, then h2 staging
    __shared__ float Wt[128][132];   // W^T: Wt[n][k]

    const int tid   = threadIdx.x;
    const int row0  = blockIdx.x * 128;
    const int wave  = tid >> 5;
    const int lane  = tid & 31;
    const int l16   = lane & 15;
    const int lhalf = lane >> 4;     // K-pair select (A/B), row-half (C/D)
    const int m0    = wave * 16;

    // ---- stage A tile (zero-padded) and W1^T ----
#pragma unroll
    for (int i = 0; i < 16; ++i) {
        int idx = tid + i * 256;
        int r   = idx >> 5;          // 0..127
        int c4  = idx & 31;          // float4 column
        int gr  = row0 + r;
        float4 va = make_float4(0.f, 0.f, 0.f, 0.f);
        if (gr < nrows) va = ((const float4*)(A + (size_t)gr * DFEAT))[c4];
        *(float4*)&As[r][c4 * 4] = va;
        float4 w = ((const float4*)(W1 + (size_t)r * DFEAT))[c4];
        Wt[c4 * 4 + 0][r] = w.x;
        Wt[c4 * 4 + 1][r] = w.y;
        Wt[c4 * 4 + 2][r] = w.z;
        Wt[c4 * 4 + 3][r] = w.w;
    }
    // ---- warm W2 into cache (global_prefetch_b8: no VGPRs, no LOADcnt) ----
    // 64KB = 1024 cachelines; 256 threads x 4 iters, one line (16 floats) each
#pragma unroll
    for (int j = 0; j < 4; ++j) {
        __builtin_prefetch(W2 + (size_t)(j * 256 + tid) * 16, 0, 3);
    }
    __syncthreads();

    // ---- GEMM1: acc = A @ W1 ----
    v8f acc[8];
#pragma unroll
    for (int nt = 0; nt < 8; ++nt) { v8f z = {}; acc[nt] = z; }
    for (int k = 0; k < 128; k += 4) {
        const int ka = k + lhalf * 2;
        v2f a = *(const v2f*)&As[m0 + l16][ka];
#pragma unroll
        for (int nt = 0; nt < 8; ++nt) {
            v2f b = *(const v2f*)&Wt[nt * 16 + l16][ka];
            acc[nt] = __builtin_amdgcn_wmma_f32_16x16x4_f32(
                false, a, false, b, (short)0, acc[nt], false, false);
        }
    }
    __syncthreads();   // all GEMM1 reads of As/Wt complete

    // ---- h1 = relu(acc + b1) back into As; reload Wt with W2^T ----
#pragma unroll
    for (int nt = 0; nt < 8; ++nt) {
        const int   n  = nt * 16 + l16;
        const float bn = b1[n];
#pragma unroll
        for (int r = 0; r < 8; ++r) {
            As[m0 + lhalf * 8 + r][n] = fmaxf(acc[nt][r] + bn, 0.0f);
        }
    }
#pragma unroll
    for (int i = 0; i < 16; ++i) {
        int idx = tid + i * 256;
        int r   = idx >> 5;
        int c4  = idx & 31;
        float4 w = ((const float4*)(W2 + (size_t)r * DFEAT))[c4];
        Wt[c4 * 4 + 0][r] = w.x;
        Wt[c4 * 4 + 1][r] = w.y;
        Wt[c4 * 4 + 2][r] = w.z;
        Wt[c4 * 4 + 3][r] = w.w;
    }
    __syncthreads();

    // ---- GEMM2: acc = h1 @ W2 ----
#pragma unroll
    for (int nt = 0; nt < 8; ++nt) { v8f z = {}; acc[nt] = z; }
    for (int k = 0; k < 128; k += 4) {
        const int ka = k + lhalf * 2;
        v2f a = *(const v2f*)&As[m0 + l16][ka];
#pragma unroll
        for (int nt = 0; nt < 8; ++nt) {
            v2f b = *(const v2f*)&Wt[nt * 16 + l16][ka];
            acc[nt] = __builtin_amdgcn_wmma_f32_16x16x4_f32(
                false, a, false, b, (short)0, acc[nt], false, false);
        }
    }
    __syncthreads();   // all GEMM2 reads of As complete

    // ---- epilogue: stage (acc + b2) in LDS, coalesced float4 stores ----
#pragma unroll
    for (int nt = 0; nt < 8; ++nt) {
        const int   n  = nt * 16 + l16;
        const float bn = b2[n];
#pragma unroll
        for (int r = 0; r < 8; ++r) {
            As[m0 + lhalf * 8 + r][n] = acc[nt][r] + bn;
        }
    }
    __syncthreads();
#pragma unroll
    for (int i = 0; i < 16; ++i) {
        int idx = tid + i * 256;
        int r   = idx >> 5;
        int c4  = idx & 31;
        int gr  = row0 + r;
        if (gr < nrows) {
            ((float4*)(Out + (size_t)gr * DFEAT))[c4] = *(const float4*)&As[r][c4 * 4];
        }
    }
}

// ---------------------------------------------------------------------------
// Kernel 4: per-feature sum / sumsq over the node dim (128-row block tiles).
// ---------------------------------------------------------------------------
__global__ __launch_bounds__(256)
void gin_stats_kernel(const float* __restrict__ h, float* __restrict__ stats,
                      int nrows) {
    __shared__ float ssum[256];
    __shared__ float ssq[256];
    const int c    = threadIdx.x & 127;
    const int half = threadIdx.x >> 7;      // 0 or 1
    const int row0 = blockIdx.x * 128 + half * 64;
    float s = 0.f, q = 0.f;
    for (int r = 0; r < 64; ++r) {
        int gr = row0 + r;
        if (gr < nrows) {
            float v = h[(size_t)gr * DFEAT + c];
            s += v;
            q += v * v;
        }
    }
    ssum[threadIdx.x] = s;
    ssq[threadIdx.x]  = q;
    __syncthreads();
    if (half == 0) {
        s += ssum[128 + c];
        q += ssq[128 + c];
        unsafeAtomicAdd(&stats[c], s);
        unsafeAtomicAdd(&stats[128 + c], q);
    }
}

// ---------------------------------------------------------------------------
// Kernel 5: BN (batch stats) + residual: out = (h-mean)*rsqrt(var+eps)*g+b + x
// ---------------------------------------------------------------------------
__global__ __launch_bounds__(256)
void gin_bn_kernel(const float* __restrict__ h, const float* __restrict__ x,
                   const float* __restrict__ stats,
                   const float* __restrict__ gamma,
                   const float* __restrict__ beta,
                   float* __restrict__ out, int total) {
    int i = blockIdx.x * 256 + threadIdx.x;
    if (i >= total) return;
    const int   c    = i & 127;
    const float invN = 1.0f / (float)NNODES;
    const float mean = stats[c] * invN;
    const float var  = stats[128 + c] * invN - mean * mean;
    const float sc   = rsqrtf(var + BN_EPS) * gamma[c];
    out[i] = (h[i] - mean) * sc + beta[c] + x[i];
}

// ---------------------------------------------------------------------------
extern "C" void kernel_launch(void* const* d_in, const int* in_sizes, int n_in,
                              void* d_out, int out_size, void* d_ws, size_t ws_size,
                              hipStream_t stream) {
    const float* x     = (const float*)d_in[0];
    const int*   eidx  = (const int*)d_in[1];      // [2, NEDGES]
    const float* W1    = (const float*)d_in[2];
    const float* b1    = (const float*)d_in[3];
    const float* W2    = (const float*)d_in[4];
    const float* b2    = (const float*)d_in[5];
    const float* gamma = (const float*)d_in[6];
    const float* beta  = (const float*)d_in[7];
    float*       out   = (float*)d_out;

    const int* src = eidx;
    const int* dst = eidx + NEDGES;

    const size_t rows_pad = 50048;                 // 391 * 128
    float* buf0  = (float*)d_ws;                   // aggr
    float* buf1  = buf0 + rows_pad * DFEAT;        // h2
    float* stats = buf1 + rows_pad * DFEAT;        // 256 floats

    const int n4       = NNODES * DFEAT / 4;       // 1.6M float4
    const int gInit    = (n4 + 255) / 256;
    const int gScatter = (int)(((size_t)NEDGES * 32 + 255) / 256);
    const int gGemm    = (NNODES + 127) / 128;     // 391
    const int total    = NNODES * DFEAT;
    const int gBn      = (total + 255) / 256;

    // 1) aggr = x ; stats = 0
    gin_init_kernel<<<gInit, 256, 0, stream>>>(x, buf0, stats, n4);
    // 2) aggr[dst] += x[src]
    gin_scatter_kernel<<<gScatter, 256, 0, stream>>>(x, src, dst, buf0, NEDGES);
    // 3) h2 = relu(aggr @ W1 + b1) @ W2 + b2   (fused, single pass)
    gin_mlp_fused_kernel<<<gGemm, 256, 0, stream>>>(buf0, W1, b1, W2, b2, buf1, NNODES);
    // 4) per-feature sum / sumsq
    gin_stats_kernel<<<gGemm, 256, 0, stream>>>(buf1, stats, NNODES);
    // 5) BN + residual
    gin_bn_kernel<<<gBn, 256, 0, stream>>>(buf1, x, stats, gamma, beta, out, total);
}